// _GCNLayer_87385404604759
// MI455X (gfx1250) — compile-verified
//
#include <hip/hip_runtime.h>
#include <math.h>

typedef __attribute__((ext_vector_type(2))) float v2f;
typedef __attribute__((ext_vector_type(8))) float v8f;

#define NN   1024
#define CIN  256
#define COUT 256
#define BATCH 32

// ---- workspace layout (in floats, all offsets even -> 8B aligned) ----
static constexpr size_t OFF_ANORM  = 0;                       // 1024*1024
static constexpr size_t OFF_APNORM = OFF_ANORM  + 1048576;    // 1024*1024
static constexpr size_t OFF_U      = OFF_APNORM + 1048576;    // 1024*16
static constexpr size_t OFF_V      = OFF_U      + 16384;      // 1024*16
static constexpr size_t OFF_DINV   = OFF_V      + 16384;      // 1024
static constexpr size_t OFF_ACC    = OFF_DINV   + 1024;       // 2 (+pad)
static constexpr size_t OFF_Y1     = OFF_ACC    + 16;         // 32*1024*256
static constexpr size_t OFF_Y2     = OFF_Y1     + 8388608;    // 32*1024*256

// ------------------------------------------------------------------
// K0: zero the elbo accumulators
__global__ void gcn_init_acc(float* acc) {
    if (threadIdx.x < 2) acc[threadIdx.x] = 0.0f;
}

// ------------------------------------------------------------------
// K1: degree of A = max(phy, phy^T, I); dinv[m] = rsqrt(rowsum)
__global__ __launch_bounds__(256) void gcn_rowsum(const float* __restrict__ phy,
                                                  float* __restrict__ dinv) {
    const int m = blockIdx.x, tid = threadIdx.x;
    float s = 0.0f;
    #pragma unroll
    for (int i = 0; i < 4; ++i) {
        const int n = tid + 256 * i;
        float a = fmaxf(phy[m * NN + n], phy[n * NN + m]);
        if (n == m) a = fmaxf(a, 1.0f);
        s += a;
    }
    __shared__ float sb[256];
    sb[tid] = s; __syncthreads();
    for (int st = 128; st > 0; st >>= 1) {
        if (tid < st) sb[tid] += sb[tid + st];
        __syncthreads();
    }
    if (tid == 0) dinv[m] = rsqrtf(sb[0]);
}

// K1b: A_norm[m][n] = dinv[m]*dinv[n]*max(phy,phy^T,I)
__global__ __launch_bounds__(256) void gcn_build_anorm(const float* __restrict__ phy,
                                                       const float* __restrict__ dinv,
                                                       float* __restrict__ An) {
    const int m = blockIdx.x, tid = threadIdx.x;
    const float dm = dinv[m];
    #pragma unroll
    for (int i = 0; i < 4; ++i) {
        const int n = tid + 256 * i;
        float a = fmaxf(phy[m * NN + n], phy[n * NN + m]);
        if (n == m) a = fmaxf(a, 1.0f);
        An[m * NN + n] = dm * dinv[n] * a;
    }
}

// ------------------------------------------------------------------
// K2a: U = emb@Wu, V = emb@Wv   (1024x64 @ 64x16, tiny)
__global__ __launch_bounds__(256) void gcn_uv(const float* __restrict__ emb,
                                              const float* __restrict__ Wu,
                                              const float* __restrict__ Wv,
                                              float* __restrict__ U, float* __restrict__ V) {
    const int gid = blockIdx.x * 256 + threadIdx.x;   // 0..32767
    const int which = gid >> 14;
    const int rem = gid & 16383;
    const int n = rem >> 4, r = rem & 15;
    const float* W = which ? Wv : Wu;
    float s = 0.0f;
    #pragma unroll 4
    for (int k = 0; k < 64; ++k) s += emb[n * 64 + k] * W[k * 16 + r];
    (which ? V : U)[n * 16 + r] = s;
}

// K2b: per-row logits -> p; Ap_norm row; elbo partial sums
__global__ __launch_bounds__(256) void gcn_pgrow(const float* __restrict__ U,
                                                 const float* __restrict__ V,
                                                 const float* __restrict__ phy,
                                                 float* __restrict__ Apn,
                                                 float* __restrict__ acc) {
    const int m = blockIdx.x, tid = threadIdx.x;
    float u[16];
    #pragma unroll
    for (int r = 0; r < 16; ++r) u[r] = U[m * 16 + r];

    const float LOG_HALF = -0.69314718055994530942f;
    float pv[4];
    float sp = 0.0f, sr = 0.0f, sk = 0.0f;
    #pragma unroll
    for (int i = 0; i < 4; ++i) {
        const int n = tid + 256 * i;
        const float* vr = V + n * 16;
        float lg = 0.0f;
        #pragma unroll
        for (int r = 0; r < 16; ++r) lg += u[r] * vr[r];
        lg *= 0.25f;                      // 1/sqrt(RANK)
        float lp, l1mp;                   // log_sigmoid(lg), log_sigmoid(-lg)
        if (lg >= 0.0f) { float e = __expf(-lg); float l = log1pf(e); lp = -l;      l1mp = -lg - l; }
        else            { float e = __expf(lg);  float l = log1pf(e); lp = lg - l;  l1mp = -l; }
        const float p = __expf(lp);       // sigmoid(lg)
        const float ph = fmaxf(phy[m * NN + n], phy[n * NN + m]);
        sr += ph * lp + (1.0f - ph) * l1mp;
        sk += p * (lp - LOG_HALF) + (1.0f - p) * (l1mp - LOG_HALF);
        sp += p;
        pv[i] = p + ((n == m) ? 1.0f : 0.0f);
    }
    __shared__ float sb[256];
    sb[tid] = sp; __syncthreads();
    for (int st = 128; st > 0; st >>= 1) { if (tid < st) sb[tid] += sb[tid + st]; __syncthreads(); }
    const float inv = 1.0f / (sb[0] + 1.0f);   // +1 from the identity term
    #pragma unroll
    for (int i = 0; i < 4; ++i) Apn[m * NN + tid + 256 * i] = pv[i] * inv;
    __syncthreads();
    sb[tid] = sr; __syncthreads();
    for (int st = 128; st > 0; st >>= 1) { if (tid < st) sb[tid] += sb[tid + st]; __syncthreads(); }
    if (tid == 0) atomicAdd(&acc[0], sb[0]);
    __syncthreads();
    sb[tid] = sk; __syncthreads();
    for (int st = 128; st > 0; st >>= 1) { if (tid < st) sb[tid] += sb[tid + st]; __syncthreads(); }
    if (tid == 0) atomicAdd(&acc[1], sb[0]);
}

// ------------------------------------------------------------------
// K3: Y1[b] = A_norm @ x[b] ; Y2[b] = Ap_norm @ x[b]  (shared x loads)
// grid (4 n-tiles, 8 m-tiles, 32 batches), 8 waves/block, wave tile 16x64 (x2)
__global__ __launch_bounds__(256) void gcn_spmm_dual(const float* __restrict__ An,
                                                     const float* __restrict__ Apn,
                                                     const float* __restrict__ x,
                                                     float* __restrict__ Y1,
                                                     float* __restrict__ Y2) {
    const int lane = threadIdx.x & 31;
    const int wave = threadIdx.x >> 5;
    const int half = lane >> 4;        // 0: lanes 0-15, 1: lanes 16-31
    const int rlo  = lane & 15;
    const int n0 = blockIdx.x * 64;
    const int m0 = blockIdx.y * 128 + wave * 16;
    const int b  = blockIdx.z;
    const float* xb = x + (size_t)b * (NN * CIN);

    const v8f zero = {0.f,0.f,0.f,0.f,0.f,0.f,0.f,0.f};
    v8f acc1[4], acc2[4];
    #pragma unroll
    for (int t = 0; t < 4; ++t) { acc1[t] = zero; acc2[t] = zero; }

    const int arow = (m0 + rlo) * NN;
    for (int k = 0; k < NN; k += 4) {
        // A fragments: lane holds A[m0+rlo][k + 2*half + {0,1}]
        const v2f a1 = *(const v2f*)(An  + arow + k + 2 * half);
        const v2f a2 = *(const v2f*)(Apn + arow + k + 2 * half);
        const float* xk = xb + (size_t)(k + 2 * half) * CIN;
        #pragma unroll
        for (int t = 0; t < 4; ++t) {
            const int n = n0 + 16 * t + rlo;
            v2f bf;                              // B[k + 2*half + v][n]
            bf.x = xk[n];
            bf.y = xk[CIN + n];
            acc1[t] = __builtin_amdgcn_wmma_f32_16x16x4_f32(false, a1, false, bf,
                                                            (short)0, acc1[t], false, false);
            acc2[t] = __builtin_amdgcn_wmma_f32_16x16x4_f32(false, a2, false, bf,
                                                            (short)0, acc2[t], false, false);
        }
    }
    const size_t base = ((size_t)b * NN + m0) * COUT;
    #pragma unroll
    for (int t = 0; t < 4; ++t) {
        const int n = n0 + 16 * t + rlo;
        #pragma unroll
        for (int j = 0; j < 8; ++j) {
            const int m = j + 8 * half;
            Y1[base + (size_t)m * COUT + n] = acc1[t][j];
            Y2[base + (size_t)m * COUT + n] = acc2[t][j];
        }
    }
}

// ------------------------------------------------------------------
// K4: out = (Y1@W_ag + b_ag) + sigmoid(mem@Wm) * (Y2@W_pg + b_pg)
// M = B*N = 32768 flattened rows; grid (4 n-tiles, 256 m-tiles)
__global__ __launch_bounds__(256) void gcn_outmm(const float* __restrict__ Y1,
                                                 const float* __restrict__ Y2,
                                                 const float* __restrict__ mem,
                                                 const float* __restrict__ Wag,
                                                 const float* __restrict__ Wpg,
                                                 const float* __restrict__ Wm,
                                                 const float* __restrict__ bag,
                                                 const float* __restrict__ bpg,
                                                 float* __restrict__ out) {
    const int lane = threadIdx.x & 31;
    const int wave = threadIdx.x >> 5;
    const int half = lane >> 4;
    const int rlo  = lane & 15;
    const int n0 = blockIdx.x * 64;
    const int m0 = blockIdx.y * 128 + wave * 16;

    const v8f zero = {0.f,0.f,0.f,0.f,0.f,0.f,0.f,0.f};
    v8f g1[4], g2[4], g3[4];
    #pragma unroll
    for (int t = 0; t < 4; ++t) { g1[t] = zero; g2[t] = zero; g3[t] = zero; }

    const int arow = (m0 + rlo) * CIN;
    for (int k = 0; k < CIN; k += 4) {
        const v2f a1 = *(const v2f*)(Y1  + arow + k + 2 * half);
        const v2f a2 = *(const v2f*)(Y2  + arow + k + 2 * half);
        const v2f a3 = *(const v2f*)(mem + arow + k + 2 * half);
        const int krow = (k + 2 * half) * COUT;
        #pragma unroll
        for (int t = 0; t < 4; ++t) {
            const int n = n0 + 16 * t + rlo;
            v2f b1, b2, b3;
            b1.x = Wag[krow + n];        b1.y = Wag[krow + COUT + n];
            b2.x = Wpg[krow + n];        b2.y = Wpg[krow + COUT + n];
            b3.x = Wm [krow + n];        b3.y = Wm [krow + COUT + n];
            g1[t] = __builtin_amdgcn_wmma_f32_16x16x4_f32(false, a1, false, b1,
                                                          (short)0, g1[t], false, false);
            g2[t] = __builtin_amdgcn_wmma_f32_16x16x4_f32(false, a2, false, b2,
                                                          (short)0, g2[t], false, false);
            g3[t] = __builtin_amdgcn_wmma_f32_16x16x4_f32(false, a3, false, b3,
                                                          (short)0, g3[t], false, false);
        }
    }
    #pragma unroll
    for (int t = 0; t < 4; ++t) {
        const int n = n0 + 16 * t + rlo;
        const float ba = bag[n], bp = bpg[n];
        #pragma unroll
        for (int j = 0; j < 8; ++j) {
            const int row = m0 + j + 8 * half;
            const float gate = 1.0f / (1.0f + __expf(-g3[t][j]));
            out[(size_t)row * COUT + n] = (g1[t][j] + ba) + (g2[t][j] + bp) * gate;
        }
    }
}

// ------------------------------------------------------------------
// K5: elbo = mean(recon) - mean(kl)
__global__ void gcn_elbo(const float* __restrict__ acc, float* __restrict__ out_elbo) {
    const float invNN = 1.0f / ((float)NN * (float)NN);
    *out_elbo = acc[0] * invNN - acc[1] * invNN;
}

// ------------------------------------------------------------------
extern "C" void kernel_launch(void* const* d_in, const int* in_sizes, int n_in,
                              void* d_out, int out_size, void* d_ws, size_t ws_size,
                              hipStream_t stream) {
    (void)in_sizes; (void)n_in; (void)out_size; (void)ws_size;
    const float* x      = (const float*)d_in[0];
    const float* memory = (const float*)d_in[1];
    const float* phy    = (const float*)d_in[2];
    const float* W_ag   = (const float*)d_in[3];
    const float* b_ag   = (const float*)d_in[4];
    const float* emb    = (const float*)d_in[5];
    const float* Wu     = (const float*)d_in[6];
    const float* Wv     = (const float*)d_in[7];
    const float* W_pg   = (const float*)d_in[8];
    const float* b_pg   = (const float*)d_in[9];
    const float* Wm     = (const float*)d_in[10];
    float* out = (float*)d_out;
    float* ws  = (float*)d_ws;

    float* An   = ws + OFF_ANORM;
    float* Apn  = ws + OFF_APNORM;
    float* U    = ws + OFF_U;
    float* V    = ws + OFF_V;
    float* dinv = ws + OFF_DINV;
    float* acc  = ws + OFF_ACC;
    float* Y1   = ws + OFF_Y1;
    float* Y2   = ws + OFF_Y2;

    gcn_init_acc<<<1, 32, 0, stream>>>(acc);
    gcn_rowsum<<<NN, 256, 0, stream>>>(phy, dinv);
    gcn_build_anorm<<<NN, 256, 0, stream>>>(phy, dinv, An);
    gcn_uv<<<128, 256, 0, stream>>>(emb, Wu, Wv, U, V);
    gcn_pgrow<<<NN, 256, 0, stream>>>(U, V, phy, Apn, acc);
    gcn_spmm_dual<<<dim3(4, 8, BATCH), 256, 0, stream>>>(An, Apn, x, Y1, Y2);
    gcn_outmm<<<dim3(4, 256), 256, 0, stream>>>(Y1, Y2, memory, W_ag, W_pg, Wm,
                                                b_ag, b_pg, out);
    gcn_elbo<<<1, 1, 0, stream>>>(acc, out + (size_t)BATCH * NN * COUT);
}